// MyLayer_82789789597838
// MI455X (gfx1250) — compile-verified
//
#include <hip/hip_runtime.h>
#include <cstdint>

// out[b,i,j,z] = sum_{di,dj} K1[di,dj] * x[b, ROWS[i]+di, COLS[j]+dj, z]
// x : (32, 512, 512, 32) f32,  out : (32, 284, 205, 32) f32
// ROWS[i] = 3i (i<58) | i+114 (i<228) | 3i-340 (i<284)   -> r in [0,509], r+2 <= 511
// COLS[j] = j (j<129) | 5j-512 (j<205)                    -> c in [0,508], c+2 <= 510
// K1 = [1,2,1] (outer) [1,2,1]  -> separable stencil.
//
// Memory-bound (~0.85 FLOP/byte). One 256-thread block per (b,i) stages ONLY
// the needed columns of the 3 input rows into LDS (compacted: 359 slots/row,
// 135 KB total -> 2 workgroups per 320KB WGP) using per-lane
// global_load_async_to_lds_b128 (ASYNCcnt path), then computes the output row
// from LDS with separable [1,2,1] weights and nontemporal stores.
//
// Compacted slot map (per input row dr):
//   s < 131        : col = s                    (stride-1 region, identity)
//   s in [131,359) : t=s-131; col = 133 + 5*(t/3) + t%3   (3 slots per j>=129)
// Compute-side slot base for output col j:
//   j < 129 : sb = j        (taps sb, sb+1, sb+2, all < 131)
//   j >= 129: sb = 131 + 3*(j-129)

typedef float f4 __attribute__((ext_vector_type(4)));

#define NX_ 512
#define NY_ 512
#define NZ_ 32
#define RN_ 284
#define CN_ 205
#define SLOTS_ 359                    // used slots per row
#define SLOT_STRIDE_ 360              // padded row stride (slots)
#define F4_PER_ROW_ (SLOTS_ * 8)      // 2872 float4 staged per row
#define F4_ROW_STRIDE_ (SLOT_STRIDE_ * 8)  // 2880 float4 LDS stride per row
#define LDS_BYTES_ (3 * F4_ROW_STRIDE_ * 16)  // 138240 B = 135 KB

__device__ __forceinline__ int row_of(int i) {
    return (i < 58) ? 3 * i : ((i < 228) ? i + 114 : 3 * i - 340);
}

extern "C" __global__ __launch_bounds__(256)
void stencil_gather_async(const float* __restrict__ x, float* __restrict__ out) {
    extern __shared__ float smem[];
    const int tid = threadIdx.x;
    const int i   = blockIdx.x;      // output row index (0..283)
    const int b   = blockIdx.y;      // batch (0..31)
    const int r   = row_of(i);

    // Raw LDS byte address (low 32 bits of the flat LDS-aperture pointer).
    const unsigned lds_base = (unsigned)(uintptr_t)smem;

    // ---- Stage needed columns of 3 input rows via async B128 loads ----
    // Per row: 2872 float4; 256 threads -> 12 iterations (last partial,
    // EXEC-masked). Max 36 outstanding async ops/wave (< 63 ASYNCcnt cap).
#pragma unroll
    for (int dr = 0; dr < 3; ++dr) {
        const float* rowbase = x + (((size_t)b * NX_ + (size_t)(r + dr)) * (NY_ * NZ_));
        const unsigned lrow = lds_base + (unsigned)(dr * F4_ROW_STRIDE_ * 16);
        for (int f = tid; f < F4_PER_ROW_; f += 256) {
            const int slot = f >> 3;         // 0..358
            const int q    = f & 7;          // float4 within the 32-float column
            int col;
            if (slot < 131) {
                col = slot;                   // stride-1 region
            } else {
                const int t = slot - 131;     // stride-5 region: 3 slots per j
                const int g = t / 3;
                col = 133 + 5 * g + (t - 3 * g);
            }
            const unsigned va = (unsigned)(col * 128 + q * 16);  // global byte off
            const unsigned la = lrow + (unsigned)(f * 16);       // LDS byte addr
            asm volatile("global_load_async_to_lds_b128 %0, %1, %2"
                         :: "v"(la), "v"(va), "s"(rowbase)
                         : "memory");
        }
    }
    asm volatile("s_wait_asynccnt 0" ::: "memory");
    __syncthreads();

    // ---- Compute 205*32 outputs (1640 float4) from LDS ----
    const f4* sp   = (const f4*)smem;
    f4*       orow = (f4*)(out + (((size_t)b * RN_ + (size_t)i) * CN_) * NZ_);

    for (int o4 = tid; o4 < (CN_ * NZ_) / 4; o4 += 256) {
        const int j  = o4 >> 3;   // output column
        const int q  = o4 & 7;    // float4 within z
        const int sb = (j < 129) ? j : 131 + 3 * (j - 129);
        f4 acc;
#pragma unroll
        for (int dr = 0; dr < 3; ++dr) {
            const int base = dr * F4_ROW_STRIDE_ + sb * 8 + q;
            const f4 a = sp[base];        // tap c
            const f4 m = sp[base + 8];    // tap c+1
            const f4 e = sp[base + 16];   // tap c+2
            const f4 t = a + m + m + e;   // horizontal [1,2,1]
            if (dr == 0)      acc = t;
            else if (dr == 1) acc += t + t;  // vertical weight 2
            else              acc += t;
        }
        // Output is write-once: nontemporal store keeps L2 for input reuse.
        __builtin_nontemporal_store(acc, &orow[o4]);
    }
}

extern "C" void kernel_launch(void* const* d_in, const int* in_sizes, int n_in,
                              void* d_out, int out_size, void* d_ws, size_t ws_size,
                              hipStream_t stream) {
    (void)in_sizes; (void)n_in; (void)out_size; (void)d_ws; (void)ws_size;
    const float* x   = (const float*)d_in[0];
    float*       out = (float*)d_out;

    (void)hipFuncSetAttribute(reinterpret_cast<const void*>(&stencil_gather_async),
                              hipFuncAttributeMaxDynamicSharedMemorySize, LDS_BYTES_);

    dim3 grid(RN_, 32, 1);   // (i, b)
    dim3 block(256, 1, 1);   // 8 wave32 waves
    hipLaunchKernelGGL(stencil_gather_async, grid, block, LDS_BYTES_, stream, x, out);
}